// CustomCNNLoss_34969623724366
// MI455X (gfx1250) — compile-verified
//
#include <hip/hip_runtime.h>
#include <cstdint>

#define S_    7
#define SS    49
#define NB_   2
#define NC_   20
#define NOBJ_ 32
#define D_    30          // NB*5 + NC
#define LC_   5.0f
#define LN_   0.5f
#define BATCH_ 16384
#define PRED_FLOATS (SS * D_)       // 1470 floats = 5880 B per image

typedef unsigned int v4u __attribute__((ext_vector_type(4)));
typedef int          v4i __attribute__((ext_vector_type(4)));
typedef int          v8i __attribute__((ext_vector_type(8)));

// ---- CDNA5 async global->LDS DMA helpers (ASYNCcnt path) -------------------
// LDS destination address: low 32 bits of a flat shared pointer are the
// wave-relative LDS offset (flat LDS aperture = {SHARED_BASE, lds_off}).
__device__ __forceinline__ uint32_t lds_off_of(const void* p) {
    return (uint32_t)(uintptr_t)p;
}
__device__ __forceinline__ void async_ld_b64(uint32_t lds, uint64_t gaddr) {
    asm volatile("global_load_async_to_lds_b64 %0, %1, off"
                 :: "v"(lds), "v"(gaddr) : "memory");
}
__device__ __forceinline__ void async_ld_b128(uint32_t lds, uint64_t gaddr) {
    asm volatile("global_load_async_to_lds_b128 %0, %1, off"
                 :: "v"(lds), "v"(gaddr) : "memory");
}
__device__ __forceinline__ void wait_async0() {
    asm volatile("s_wait_asynccnt 0" ::: "memory");
}

// ---- CDNA5 Tensor Data Mover: one-shot DMA of the whole prediction tile ----
// D# per CDNA5 ISA: 2-D tensor, data_size=4B,
// tensor_dim0 = tile_dim0 = 1470 elems, tensor_dim1 = tile_dim1 = 1 row,
// tensor_dim0_stride = 1470, no pad/iterate/gather, workgroup_mask = 0
// (not in a cluster), groups 2/3 unused (zero).
__device__ __forceinline__ void tdm_load_pred_tile(uint32_t lds_addr, uint64_t gaddr) {
#if __has_builtin(__builtin_amdgcn_tensor_load_to_lds)
    v4u g0;
    g0[0] = 0x1u;                                    // count=1, is_restore=0, no gather
    g0[1] = lds_addr;                                // lds_addr[31:0]
    g0[2] = (uint32_t)gaddr;                         // global_addr[31:0]
    g0[3] = ((uint32_t)(gaddr >> 32) & 0x01FFFFFFu)  // global_addr[56:32]
            | 0x80000000u;                           // type=2 ("image") in [127:126]
    v8i g1;
    g1[0] = (int)(2u << 16);                         // wg_mask=0, data_size=2 (4B)
    g1[1] = (int)((uint32_t)PRED_FLOATS << 16);      // tensor_dim0 lo16 @ [31:16]
    g1[2] = (int)(1u << 16);                         // tensor_dim0 hi16=0; tensor_dim1=1
    g1[3] = (int)((uint32_t)PRED_FLOATS << 16);      // tensor_dim1 hi16=0; tile_dim0=1470
    g1[4] = 1;                                       // tile_dim1=1, tile_dim2=0
    g1[5] = PRED_FLOATS;                             // tensor_dim0_stride lo32
    g1[6] = 0;                                       // stride hi16; dim1_stride lo16
    g1[7] = 0;
    v4i gz = {0, 0, 0, 0};
#if __clang_major__ >= 23
    v8i gz8 = {0, 0, 0, 0, 0, 0, 0, 0};
    __builtin_amdgcn_tensor_load_to_lds(g0, g1, gz, gz, gz8, 0);
#else
    __builtin_amdgcn_tensor_load_to_lds(g0, g1, gz, gz, 0);
#endif
#else
    // Fallback: per-lane async DMA (5880 B = 735 x b64), lane = active lane id.
    // Caller invokes this from all 128 threads in that configuration instead.
    (void)lds_addr; (void)gaddr;
#endif
}
__device__ __forceinline__ void wait_tensor0() {
#if __has_builtin(__builtin_amdgcn_s_wait_tensorcnt)
    __builtin_amdgcn_s_wait_tensorcnt(0);
#else
    asm volatile("s_wait_tensorcnt 0" ::: "memory");
#endif
}

// ---------------------------------------------------------------------------
// Stage 1: one 128-thread block per image. TDM-stage the 5880 B pred tile,
// async-stage the 640 B targets, compute per-image loss, write partial.
// ---------------------------------------------------------------------------
__global__ __launch_bounds__(128)
void yolo_loss_stage1(const float* __restrict__ preds,
                      const float* __restrict__ tgts,
                      float* __restrict__ partial,
                      float* __restrict__ out_atomic,
                      int use_atomic) {
    __shared__ __align__(16) float sp[PRED_FLOATS];  // 1470 f = 5880 B
    __shared__ __align__(16) float st[NOBJ_ * 5];    //  160 f =  640 B
    __shared__ float t_iou[NOBJ_];
    __shared__ int   t_cell[NOBJ_];
    __shared__ int   t_bj[NOBJ_];
    __shared__ int   t_cls[NOBJ_];
    __shared__ float t_box[NOBJ_][4];
    __shared__ float csum[SS];

    const int b   = blockIdx.x;
    const int tid = threadIdx.x;

    const uint64_t pbase = (uint64_t)(uintptr_t)preds + (uint64_t)b * (PRED_FLOATS * 4);
    const uint64_t tbase = (uint64_t)(uintptr_t)tgts  + (uint64_t)b * (NOBJ_ * 5 * 4);
    const uint32_t sp_l  = lds_off_of(&sp[0]);
    const uint32_t st_l  = lds_off_of(&st[0]);

#if __has_builtin(__builtin_amdgcn_tensor_load_to_lds)
    // One TDM DMA for the whole prediction tile; issue from wave 0 only
    // (TDM ignores EXEC, so other waves must branch around it entirely).
    if (tid < 32) {
        if (tid == 0) tdm_load_pred_tile(sp_l, pbase);
    }
#else
    // Fallback: per-lane async DMA, 735 x b64 (image stride is 8B-aligned only).
    for (int i = tid; i < 735; i += 128)
        async_ld_b64(sp_l + 8u * (uint32_t)i, pbase + 8ull * (uint64_t)i);
#endif
    // Targets tile: 640 B = 40 x b128 async loads (16B-aligned per image).
    if (tid < 40)
        async_ld_b128(st_l + 16u * (uint32_t)tid, tbase + 16ull * (uint64_t)tid);

    wait_async0();
#if __has_builtin(__builtin_amdgcn_tensor_load_to_lds)
    if (tid < 32) wait_tensor0();
#endif
    __syncthreads();

    // ---- Phase A: per-target (lanes 0..31) ----
    if (tid < NOBJ_) {
        const int t  = tid;
        const float cls = st[t * 5 + 0];
        const float cx  = st[t * 5 + 1];
        const float cy  = st[t * 5 + 2];
        const float w   = st[t * 5 + 3];
        const float h   = st[t * 5 + 4];
        const bool valid = (cls >= 0.0f);

        int col = (int)floorf(cx * (float)S_);
        int row = (int)floorf(cy * (float)S_);
        col = col < 0 ? 0 : (col > S_ - 1 ? S_ - 1 : col);
        row = row < 0 ? 0 : (row > S_ - 1 ? S_ - 1 : row);
        const int cell = row * S_ + col;

        const float gx1 = cx - w * 0.5f, gy1 = cy - h * 0.5f;
        const float gx2 = cx + w * 0.5f, gy2 = cy + h * 0.5f;
        const float garea = (gx2 - gx1) * (gy2 - gy1);

        float best = -1.0f; int bj = 0;
        #pragma unroll
        for (int j = 0; j < NB_; ++j) {
            const float* pbx = &sp[cell * D_ + j * 5];
            const float pcx = pbx[0], pcy = pbx[1], pw = pbx[2], ph = pbx[3];
            const float px1 = pcx - pw * 0.5f, py1 = pcy - ph * 0.5f;
            const float px2 = pcx + pw * 0.5f, py2 = pcy + ph * 0.5f;
            float iw = fminf(px2, gx2) - fmaxf(px1, gx1); iw = fmaxf(iw, 0.0f);
            float ih = fminf(py2, gy2) - fmaxf(py1, gy1); ih = fmaxf(ih, 0.0f);
            const float inter = iw * ih;
            const float uni = (px2 - px1) * (py2 - py1) + garea - inter;
            const float iou = inter / (uni + 1e-6f);
            if (iou > best) { best = iou; bj = j; }   // tie -> lower j (argmax)
        }
        t_iou[t]  = best;
        t_bj[t]   = bj;
        t_cell[t] = valid ? cell : -1;
        float cf = cls; cf = cf < 0.0f ? 0.0f : (cf > (float)(NC_ - 1) ? (float)(NC_ - 1) : cf);
        t_cls[t] = (int)cf;
        t_box[t][0] = cx * (float)S_ - (float)col;
        t_box[t][1] = cy * (float)S_ - (float)row;
        t_box[t][2] = w;
        t_box[t][3] = h;
    }
    __syncthreads();

    // ---- Phase B: per-cell (lanes 0..48) ----
    if (tid < SS) {
        const int c = tid;
        int winner = -1; float wmax = -1.0f; unsigned mask = 0u;
        #pragma unroll 4
        for (int t = 0; t < NOBJ_; ++t) {
            if (t_cell[t] == c) {
                mask |= 1u << t_cls[t];                         // scatter .set == OR
                if (t_iou[t] > wmax) { wmax = t_iou[t]; winner = t; } // first max wins
            }
        }
        const float* p = &sp[c * D_];
        float sum;
        if (winner >= 0) {
            const int j = t_bj[winner];
            const float* pb = p + j * 5;
            const float gx = t_box[winner][0], gy = t_box[winner][1];
            const float gw = t_box[winner][2], gh = t_box[winner][3];
            const float dx = pb[0] - gx, dy = pb[1] - gy;
            const float lxy = dx * dx + dy * dy;
            const float sw = sqrtf(fmaxf(pb[2], 0.0f)) - sqrtf(gw);
            const float sh = sqrtf(fmaxf(pb[3], 0.0f)) - sqrtf(gh);
            const float lwh = sw * sw + sh * sh;
            const float dob = pb[4] - t_iou[winner];
            const float lobj = dob * dob;
            const float co = p[(1 - j) * 5 + 4];                // non-responsible box conf
            const float lnoobj = co * co;
            float lcls = 0.0f;
            #pragma unroll
            for (int k = 0; k < NC_; ++k) {
                const float g = (float)((mask >> k) & 1u);
                const float d = p[NB_ * 5 + k] - g;
                lcls += d * d;
            }
            sum = LC_ * (lxy + lwh) + lobj + LN_ * lnoobj + lcls;
        } else {
            const float c0 = p[4], c1 = p[9];                   // both boxes are "noobj"
            sum = LN_ * (c0 * c0 + c1 * c1);
        }
        csum[c] = sum;
    }
    __syncthreads();

    if (tid == 0) {
        float acc = 0.0f;
        #pragma unroll
        for (int c = 0; c < SS; ++c) acc += csum[c];
        if (use_atomic) atomicAdd(out_atomic, acc * (1.0f / (float)BATCH_));
        else            partial[b] = acc;
    }
}

// ---------------------------------------------------------------------------
// Stage 2: deterministic tree reduction of 16384 per-image partials.
// ---------------------------------------------------------------------------
__global__ __launch_bounds__(256)
void yolo_loss_reduce(const float* __restrict__ partial, float* __restrict__ out) {
    __shared__ float red[256];
    float acc = 0.0f;
    for (int i = threadIdx.x; i < BATCH_; i += 256) acc += partial[i];
    red[threadIdx.x] = acc;
    __syncthreads();
    for (int s = 128; s > 0; s >>= 1) {
        if ((int)threadIdx.x < s) red[threadIdx.x] += red[threadIdx.x + s];
        __syncthreads();
    }
    if (threadIdx.x == 0) out[0] = red[0] * (1.0f / (float)BATCH_);
}

__global__ void zero_out(float* out) { if (threadIdx.x == 0) out[0] = 0.0f; }

extern "C" void kernel_launch(void* const* d_in, const int* in_sizes, int n_in,
                              void* d_out, int out_size, void* d_ws, size_t ws_size,
                              hipStream_t stream) {
    const float* preds = (const float*)d_in[0];   // (16384,7,7,30) f32
    const float* tgts  = (const float*)d_in[1];   // (16384,32,5)   f32
    float* out = (float*)d_out;

    if (ws_size >= (size_t)BATCH_ * sizeof(float)) {
        float* partial = (float*)d_ws;
        yolo_loss_stage1<<<BATCH_, 128, 0, stream>>>(preds, tgts, partial, nullptr, 0);
        yolo_loss_reduce<<<1, 256, 0, stream>>>(partial, out);
    } else {
        // Fallback: atomic accumulation (workspace too small)
        zero_out<<<1, 32, 0, stream>>>(out);
        yolo_loss_stage1<<<BATCH_, 128, 0, stream>>>(preds, tgts, nullptr, out, 1);
    }
}